// GCNLayer_5085241279101
// MI455X (gfx1250) — compile-verified
//
#include <hip/hip_runtime.h>
#include <hip/hip_bf16.h>

#define D 128          // D_IN == D_OUT == 128
#define SCAN_T 1024    // threads in the scan kernels

typedef __attribute__((ext_vector_type(2))) float v2f;
typedef __attribute__((ext_vector_type(8))) float v8f;

// ---------------------------------------------------------------------------
// 1) zero degree array (d_ws is poisoned with 0xAA)
// ---------------------------------------------------------------------------
__global__ void gcn_zero_u32(unsigned* __restrict__ p, int n) {
    int i = blockIdx.x * blockDim.x + threadIdx.x;
    if (i < n) p[i] = 0u;
}

// ---------------------------------------------------------------------------
// 2) degree count (integer atomics: result order-independent -> deterministic)
// ---------------------------------------------------------------------------
__global__ void gcn_degree(const long long* __restrict__ rows,
                           unsigned* __restrict__ deg, int nEdges) {
    int e = blockIdx.x * blockDim.x + threadIdx.x;
    if (e < nEdges) atomicAdd(&deg[(int)rows[e]], 1u);
}

// ---------------------------------------------------------------------------
// 3) isd[i] = rsqrt(max(deg,1))
// ---------------------------------------------------------------------------
__global__ void gcn_rsqrt(const unsigned* __restrict__ deg,
                          float* __restrict__ isd, int nNodes) {
    int i = blockIdx.x * blockDim.x + threadIdx.x;
    if (i < nNodes) isd[i] = rsqrtf(fmaxf((float)deg[i], 1.0f));
}

// ---------------------------------------------------------------------------
// 4) three-step exclusive prefix sum of deg -> rowStart, cursor
// ---------------------------------------------------------------------------
__global__ void gcn_scan_part(const unsigned* __restrict__ deg,
                              unsigned* __restrict__ partials, int nNodes, int chunk) {
    int t = threadIdx.x;                       // single block of SCAN_T
    long long s = (long long)t * chunk;
    long long e = s + chunk; if (e > nNodes) e = nNodes;
    unsigned sum = 0;
    for (long long i = s; i < e; ++i) sum += deg[i];
    partials[t] = sum;
}
__global__ void gcn_scan_top(const unsigned* __restrict__ partials,
                             unsigned* __restrict__ psum) {
    if (threadIdx.x == 0) {                    // 1024 sequential adds: negligible
        unsigned run = 0;
        for (int i = 0; i < SCAN_T; ++i) { psum[i] = run; run += partials[i]; }
    }
}
__global__ void gcn_scan_write(const unsigned* __restrict__ deg,
                               const unsigned* __restrict__ psum,
                               unsigned* __restrict__ rowStart,
                               unsigned* __restrict__ cursor, int nNodes, int chunk) {
    int t = threadIdx.x;
    long long s = (long long)t * chunk;
    long long e = s + chunk; if (e > nNodes) e = nNodes;
    unsigned run = psum[t];
    for (long long i = s; i < e; ++i) {
        rowStart[i] = run;
        cursor[i]   = run;
        run += deg[i];
    }
}

// ---------------------------------------------------------------------------
// 5) CSR bucket fill: csrCol[cursor[row]++] = col
// ---------------------------------------------------------------------------
__global__ void gcn_csr_fill(const long long* __restrict__ rows,
                             const long long* __restrict__ cols,
                             unsigned* __restrict__ cursor,
                             int* __restrict__ csrCol, int nEdges) {
    int e = blockIdx.x * blockDim.x + threadIdx.x;
    if (e >= nEdges) return;
    int r = (int)rows[e];
    unsigned slot = atomicAdd(&cursor[r], 1u);
    csrCol[slot] = (int)cols[e];
}

// ---------------------------------------------------------------------------
// 6) per-node selection sort of its bucket -> fixed accumulation order
//    (avg degree 16 -> O(d^2) is cheap; makes output bit-deterministic)
// ---------------------------------------------------------------------------
__global__ void gcn_sort_buckets(int* __restrict__ csrCol,
                                 const unsigned* __restrict__ rowStart,
                                 const unsigned* __restrict__ deg, int nNodes) {
    int node = blockIdx.x * blockDim.x + threadIdx.x;
    if (node >= nNodes) return;
    unsigned s = rowStart[node], d = deg[node];
    for (unsigned i = 0; i + 1 < d; ++i) {
        unsigned mi = i; int mv = csrCol[s + i];
        for (unsigned j = i + 1; j < d; ++j) {
            int v = csrCol[s + j];
            if (v < mv) { mv = v; mi = j; }
        }
        csrCol[s + mi] = csrCol[s + i];
        csrCol[s + i]  = mv;
    }
}

// ---------------------------------------------------------------------------
// 7) fused gather + WMMA GEMM + bias.
//    Block = 256 thr = 8 waves handles 16 nodes (one 16-row M tile).
//    Gather: wave w accumulates rows 2w, 2w+1 into registers (lane owns
//    dims [4*lane,4*lane+4)), stores to padded LDS tile (stride 132 floats
//    -> A-frag ds_load_b64 is bank-conflict-free across the 16 lanes).
//    GEMM:   wave w = N-tile w; 32 chained V_WMMA_F32_16X16X4_F32 over K=128.
//    Loop bounds are wave-uniform -> EXEC is all-ones at every WMMA.
// ---------------------------------------------------------------------------
#define LDS_STRIDE 132
__global__ void gcn_fused(const float* __restrict__ x,
                          const int* __restrict__ csrCol,
                          const unsigned* __restrict__ rowStart,
                          const unsigned* __restrict__ deg,
                          const float* __restrict__ isd,
                          const float* __restrict__ W,
                          const float* __restrict__ bias,
                          float* __restrict__ out) {
    __shared__ float tile[16 * LDS_STRIDE];          // 8448 B of the 320 KB LDS
    const int lane  = threadIdx.x & 31;
    const int wv    = threadIdx.x >> 5;              // 0..7
    const int mBase = blockIdx.x * 16;               // nNodes % 16 == 0 (100000/16)

    // ---- gather: 2 node rows per wave ----
#pragma unroll
    for (int rr = 0; rr < 2; ++rr) {
        const int node = mBase + wv * 2 + rr;
        const unsigned s = rowStart[node];
        const unsigned e = s + deg[node];
        const float ir = isd[node];
        float4 acc = make_float4(0.f, 0.f, 0.f, 0.f);
        const float* xb = x + lane * 4;
        for (unsigned i = s; i < e; ++i) {           // wave-uniform trip count
            const int c = csrCol[i];
            const float nrm = ir * isd[c];
            const float4 xv = *(const float4*)(xb + (size_t)c * D);
            acc.x += xv.x * nrm; acc.y += xv.y * nrm;
            acc.z += xv.z * nrm; acc.w += xv.w * nrm;
        }
        *(float4*)(&tile[(wv * 2 + rr) * LDS_STRIDE + lane * 4]) = acc;
    }
    __syncthreads();

    // ---- WMMA: out[m][n] = sum_k tile[m][k] * W[n][k] + b[n] ----
    const int nBase = wv * 16;
    const int mn    = lane & 15;                     // A row / B col
    const int kOff  = (lane >> 4) * 2;               // 0 or 2
    const float* bPtr   = W + (size_t)(nBase + mn) * D + kOff;   // W[n][k], contig in k
    const float* aLds   = &tile[mn * LDS_STRIDE + kOff];

    v8f c = {};
#pragma unroll
    for (int k0 = 0; k0 < D; k0 += 4) {
        float2 av = *(const float2*)(aLds + k0);     // ds_load_b64, no bank conflicts
        float2 bv = *(const float2*)(bPtr + k0);
        v2f a = {av.x, av.y};
        v2f b = {bv.x, bv.y};
        c = __builtin_amdgcn_wmma_f32_16x16x4_f32(false, a, false, b,
                                                  (short)0, c, false, false);
    }

    const float bv = bias[nBase + mn];
    const int mHalf = (lane >> 4) * 8;               // C/D: vgpr r -> M = r + 8*(lane>=16)
#pragma unroll
    for (int r = 0; r < 8; ++r)
        out[(size_t)(mBase + mHalf + r) * D + nBase + mn] = c[r] + bv;
}

// ---------------------------------------------------------------------------
extern "C" void kernel_launch(void* const* d_in, const int* in_sizes, int n_in,
                              void* d_out, int out_size, void* d_ws, size_t ws_size,
                              hipStream_t stream) {
    const float*     x    = (const float*)d_in[0];
    const long long* eidx = (const long long*)d_in[1];   // int64 per reference
    const float*     W    = (const float*)d_in[2];
    const float*     bias = (const float*)d_in[3];
    float*           out  = (float*)d_out;

    const int nNodes = in_sizes[0] / D;        // 100000
    const int nEdges = in_sizes[1] / 2;        // 1600000
    const long long* rows = eidx;
    const long long* cols = eidx + nEdges;

    // workspace layout (all 4-byte types): ~8.1 MB total
    unsigned* deg      = (unsigned*)d_ws;
    float*    isd      = (float*)(deg + nNodes);
    unsigned* rowStart = (unsigned*)(isd + nNodes);
    unsigned* cursor   = rowStart + nNodes;
    unsigned* partials = cursor + nNodes;
    unsigned* psum     = partials + SCAN_T;
    int*      csrCol   = (int*)(psum + SCAN_T);

    const int chunk = (nNodes + SCAN_T - 1) / SCAN_T;

    gcn_zero_u32<<<(nNodes + 255) / 256, 256, 0, stream>>>(deg, nNodes);
    gcn_degree<<<(nEdges + 255) / 256, 256, 0, stream>>>(rows, deg, nEdges);
    gcn_rsqrt<<<(nNodes + 255) / 256, 256, 0, stream>>>(deg, isd, nNodes);
    gcn_scan_part<<<1, SCAN_T, 0, stream>>>(deg, partials, nNodes, chunk);
    gcn_scan_top<<<1, 32, 0, stream>>>(partials, psum);
    gcn_scan_write<<<1, SCAN_T, 0, stream>>>(deg, psum, rowStart, cursor, nNodes, chunk);
    gcn_csr_fill<<<(nEdges + 255) / 256, 256, 0, stream>>>(rows, cols, cursor, csrCol, nEdges);
    gcn_sort_buckets<<<(nNodes + 255) / 256, 256, 0, stream>>>(csrCol, rowStart, deg, nNodes);
    gcn_fused<<<nNodes / 16, 256, 0, stream>>>(x, csrCol, rowStart, deg, isd, W, bias, out);
}